// DetrLoss_32238024523847
// MI455X (gfx1250) — compile-verified
//
#include <hip/hip_runtime.h>
#include <hip/hip_bf16.h>
#include <math.h>

#define BB 32
#define NN 4096
#define MM 256
#define NT 1024
#define RPT (NN / NT) /* rows per thread = 4 */
#define EPSV 1e-7f

typedef float v2f  __attribute__((ext_vector_type(2)));
typedef float v8f  __attribute__((ext_vector_type(8)));
typedef unsigned int u32x4 __attribute__((ext_vector_type(4)));
typedef int i32x8 __attribute__((ext_vector_type(8)));
typedef int i32x4 __attribute__((ext_vector_type(4)));

#if defined(__has_builtin)
# if __has_builtin(__builtin_amdgcn_tensor_load_to_lds) && __has_builtin(__builtin_amdgcn_s_wait_tensorcnt)
#  define USE_TDM 1
# endif
# if __has_builtin(__builtin_amdgcn_wmma_f32_16x16x4_f32)
#  define USE_WMMA 1
# endif
#endif

__device__ __forceinline__ float iou_pair(const float4 p, float pa,
                                          const float4 g, float ga) {
  float xi1 = fmaxf(p.x, g.x), yi1 = fmaxf(p.y, g.y);
  float xi2 = fminf(p.z, g.z), yi2 = fminf(p.w, g.w);
  float inter = fmaxf(xi2 - xi1, 0.0f) * fmaxf(yi2 - yi1, 0.0f);
  float uni = pa + ga - inter;         // reference _box_iou has no eps
  return inter / uni;
}

__device__ __forceinline__ float giou_loss(const float4 p, const float4 g) {
  float xi1 = fmaxf(p.x, g.x), yi1 = fmaxf(p.y, g.y);
  float xi2 = fminf(p.z, g.z), yi2 = fminf(p.w, g.w);
  float inter = fmaxf(xi2 - xi1, 0.0f) * fmaxf(yi2 - yi1, 0.0f);
  float a1 = (p.z - p.x) * (p.w - p.y);
  float a2 = (g.z - g.x) * (g.w - g.y);
  float uni = a1 + a2 - inter;
  float iou = inter / (uni + EPSV);
  float xc1 = fminf(p.x, g.x), yc1 = fminf(p.y, g.y);
  float xc2 = fmaxf(p.z, g.z), yc2 = fmaxf(p.w, g.w);
  float ac = (xc2 - xc1) * (yc2 - yc1);
  float giou = iou - (ac - uni) / (ac + EPSV);
  return 1.0f - giou;
}

__global__ void __launch_bounds__(NT)
detr_match_loss(const float* __restrict__ pr, const float* __restrict__ gt,
                float* __restrict__ ws) {
  __shared__ float4 gtbS[MM];        // GT boxes (TDM target)
  __shared__ float  gtAreaS[MM];
  __shared__ float  rowValS[NN];     // per-row best IoU over unmasked cols
  __shared__ int    rowColS[NN];     // its argmax column
  __shared__ int    colMaskS[MM];
  __shared__ unsigned char matchedS[NN];
  __shared__ int    selRowS[MM];
  __shared__ int    selColS[MM];
  __shared__ float  redV[32];
  __shared__ int    redI[32];
  __shared__ int    bcastJ;

  const int tid  = threadIdx.x;
  const int b    = blockIdx.x;
  const int lane = tid & 31;
  const int wid  = tid >> 5;
  const float NEG_INFF = -__builtin_inff();

  // ---- DMA the 256x4 f32 GT tile into LDS via the Tensor Data Mover ----
#ifdef USE_TDM
  if (wid == 0) {
    unsigned long long ga = (unsigned long long)(gt + (size_t)b * MM * 4);
    unsigned int ldsOff = (unsigned int)(unsigned long long)(void*)&gtbS[0];
    u32x4 g0;
    g0[0] = 1u;                                   // count=1, user descriptor
    g0[1] = ldsOff;                               // lds_addr [63:32]
    g0[2] = (unsigned int)ga;                     // global_addr lo
    g0[3] = (unsigned int)((ga >> 32) & 0x1FFFFFFull) | (2u << 30); // hi + type=2
    i32x8 g1;
    g1[0] = (int)(2u << 16);                      // data_size = 4B
    g1[1] = (int)(((unsigned)(MM * 4) & 0xFFFFu) << 16); // tensor_dim0=1024 (lo16)
    g1[2] = (int)(1u << 16);                      // tensor_dim1 = 1
    g1[3] = (int)(((unsigned)(MM * 4) & 0xFFFFu) << 16); // tile_dim0 = 1024
    g1[4] = 1;                                    // tile_dim1 = 1
    g1[5] = MM * 4;                               // tensor_dim0_stride = 1024
    g1[6] = (int)(((unsigned)(MM * 4) & 0xFFFFu) << 16); // tensor_dim1_stride
    g1[7] = 0;
    i32x4 gz4 = {0, 0, 0, 0};                     // groups 2/3 unused (2D tensor)
    i32x8 gz8 = {0, 0, 0, 0, 0, 0, 0, 0};
    __builtin_amdgcn_tensor_load_to_lds(g0, g1, gz4, gz4, gz8, 0);
    __builtin_amdgcn_s_wait_tensorcnt(0);
  }
#else
  for (int j = tid; j < MM; j += NT)
    gtbS[j] = reinterpret_cast<const float4*>(gt)[(size_t)b * MM + j];
#endif
  __syncthreads();

  if (tid < MM) {
    float4 g = gtbS[tid];
    gtAreaS[tid] = (g.z - g.x) * (g.w - g.y);
    colMaskS[tid] = 0;
  }

  // ---- load this thread's 4 prediction boxes; init flags ----
  float4 pb[RPT];
  float  pa[RPT];
  const float4* pr4 = reinterpret_cast<const float4*>(pr) + (size_t)b * NN;
#pragma unroll
  for (int k = 0; k < RPT; ++k) {
    int r = tid + k * NT;
    pb[k] = pr4[r];
    pa[k] = (pb[k].z - pb[k].x) * (pb[k].w - pb[k].y);
    matchedS[r] = 0;
  }
  __syncthreads();

  // ---- initial per-row best IoU (strict '>' keeps smallest col on ties) ----
#pragma unroll
  for (int k = 0; k < RPT; ++k) {
    int r = tid + k * NT;
    float bv = NEG_INFF; int bc = 0;
    for (int j = 0; j < MM; ++j) {
      float v = iou_pair(pb[k], pa[k], gtbS[j], gtAreaS[j]);
      if (v > bv) { bv = v; bc = j; }
    }
    rowValS[r] = bv; rowColS[r] = bc;
  }
  __syncthreads();

  // ---- greedy maximal-IoU matching: 256 sequential global argmax steps ----
  for (int step = 0; step < MM; ++step) {
    float bv = NEG_INFF; int bi = 0x7fffffff;
#pragma unroll
    for (int k = 0; k < RPT; ++k) {
      int r = tid + k * NT;
      float v = rowValS[r];
      if (v > bv || (v == bv && r < bi)) { bv = v; bi = r; }
    }
    for (int off = 16; off > 0; off >>= 1) {      // wave32 argmax
      float ov = __shfl_xor(bv, off, 32);
      int   oi = __shfl_xor(bi, off, 32);
      if (ov > bv || (ov == bv && oi < bi)) { bv = ov; bi = oi; }
    }
    if (lane == 0) { redV[wid] = bv; redI[wid] = bi; }
    __syncthreads();
    if (wid == 0) {
      bv = redV[lane]; bi = redI[lane];
      for (int off = 16; off > 0; off >>= 1) {
        float ov = __shfl_xor(bv, off, 32);
        int   oi = __shfl_xor(bi, off, 32);
        if (ov > bv || (ov == bv && oi < bi)) { bv = ov; bi = oi; }
      }
      if (lane == 0) {
        int jst = rowColS[bi];
        selRowS[step] = bi; selColS[step] = jst;
        rowValS[bi] = NEG_INFF;
        matchedS[bi] = 1;
        colMaskS[jst] = 1;
        bcastJ = jst;
      }
    }
    __syncthreads();
    if (step < MM - 1) {
      int jst = bcastJ;   // only rows whose best col was just masked rescan
#pragma unroll
      for (int k = 0; k < RPT; ++k) {
        int r = tid + k * NT;
        if (rowColS[r] == jst && rowValS[r] != NEG_INFF) {
          float nv = NEG_INFF; int nc = 0;
          for (int j = 0; j < MM; ++j) {
            if (colMaskS[j]) continue;
            float v = iou_pair(pb[k], pa[k], gtbS[j], gtAreaS[j]);
            if (v > nv) { nv = v; nc = j; }
          }
          rowValS[r] = nv; rowColS[r] = nc;
        }
      }
    }
    __syncthreads();
  }

  // ---- GIoU losses: matched mean (M) + unmatched-vs-zero mean (N-M) ----
  float local = 0.0f;
  if (tid < MM) {
    int i = selRowS[tid], j = selColS[tid];
    local += giou_loss(pr4[i], gtbS[j]) * (1.0f / MM);
  }
  const float4 zero4 = make_float4(0.0f, 0.0f, 0.0f, 0.0f);
#pragma unroll
  for (int k = 0; k < RPT; ++k) {
    int r = tid + k * NT;
    if (!matchedS[r])
      local += giou_loss(pb[k], zero4) * (1.0f / (NN - MM));
  }
  for (int off = 16; off > 0; off >>= 1) local += __shfl_xor(local, off, 32);
  if (lane == 0) redV[wid] = local;
  __syncthreads();
  if (wid == 0) {
    float s = redV[lane];
    for (int off = 16; off > 0; off >>= 1) s += __shfl_xor(s, off, 32);
    if (lane == 0) ws[b] = s;
  }
}

// Single-wave finalize: sum 32 batch partials with v_wmma_f32_16x16x4_f32
// (A = partials in 16x4 layout, B = ones -> D row-sums), then /B/2.
__global__ void __launch_bounds__(32)
detr_finalize(const float* __restrict__ ws, float* __restrict__ out) {
  int lane = threadIdx.x;
  float v = ws[lane];
#ifdef USE_WMMA
  v2f a;  a[0] = v;    a[1] = 0.0f;     // lanes0-15: K=0; lanes16-31: K=2
  v2f bo; bo[0] = 1.0f; bo[1] = 1.0f;   // all-ones B (layout-independent)
  v8f c = {0.f, 0.f, 0.f, 0.f, 0.f, 0.f, 0.f, 0.f};
  c = __builtin_amdgcn_wmma_f32_16x16x4_f32(false, a, false, bo, (short)0, c,
                                            false, false);
  float s = c[0] + c[1] + c[2] + c[3] + c[4] + c[5] + c[6] + c[7];
  s += __shfl_xor(s, 16, 32);           // lane0: rows 0-7 + rows 8-15 at N=0
  if (lane == 0) out[0] = s * (1.0f / 64.0f);   // / B / 2
#else
  for (int off = 16; off > 0; off >>= 1) v += __shfl_xor(v, off, 32);
  if (lane == 0) out[0] = v * (1.0f / 64.0f);
#endif
}

extern "C" void kernel_launch(void* const* d_in, const int* in_sizes, int n_in,
                              void* d_out, int out_size, void* d_ws, size_t ws_size,
                              hipStream_t stream) {
  const float* pr = (const float*)d_in[0];   // pr_boxes [32,4096,4]
  const float* gt = (const float*)d_in[1];   // gt_boxes [32,256,4]
  // d_in[2] (logits) and d_in[3] (gt_classes) are dead code in the reference.
  float* wsf = (float*)d_ws;                 // 32 per-batch partials
  detr_match_loss<<<dim3(BB), dim3(NT), 0, stream>>>(pr, gt, wsf);
  detr_finalize<<<dim3(1), dim3(32), 0, stream>>>(wsf, (float*)d_out);
}